// ZBLRepulsion_2095944040954
// MI455X (gfx1250) — compile-verified
//
#include <hip/hip_runtime.h>
#include <hip/hip_bf16.h>
#include <math.h>

#define TILE 256                    // pairs per tile == blockDim.x
#define DISP_BYTES (TILE * 12)      // 3072 B displacements per tile
#define IDX_BYTES  (TILE * 4)       // 1024 B per int/float stream per tile
#define BUF_BYTES  (DISP_BYTES + 3 * IDX_BYTES)   // 6144 B per buffer

typedef __attribute__((ext_vector_type(4))) unsigned int v4u;
typedef __attribute__((ext_vector_type(4))) int          v4i;
typedef __attribute__((ext_vector_type(8))) int          v8i;

// ---------------------------------------------------------------------------
// TDM: issue one 1-D tensor_load_to_lds (D# per CDNA5 ISA).
//   group0: [1:0]=count=1 | [63:32]=lds_addr | [120:64]=global_addr | type=2
//   group1: wg_mask=0, data_size=2 (4B), tensor_dim0=tensor_dw, tensor_dim1=1,
//           tile_dim0=tile_dw, tile_dim1/2=0 (1-D), dim0_stride=tensor_dw
// ---------------------------------------------------------------------------
__device__ __forceinline__ void tdm_load_1d(unsigned lds_off, const void* gaddr,
                                            unsigned tile_dw, unsigned tensor_dw) {
    unsigned long long ga = (unsigned long long)gaddr;
    v4u g0;
    g0[0] = 1u;                                           // count=1, user desc
    g0[1] = lds_off;                                      // lds_addr (bytes)
    g0[2] = (unsigned)ga;                                 // global_addr[31:0]
    g0[3] = ((unsigned)(ga >> 32) & 0x01FFFFFFu)          // global_addr[56:32]
          | (2u << 30);                                   // type=2 ("image")
    v8i g1;
    g1[0] = (int)(2u << 16);                              // data_size=4B, mask=0
    g1[1] = (int)((tensor_dw & 0xFFFFu) << 16);           // tensor_dim0[15:0]
    g1[2] = (int)((tensor_dw >> 16) | (1u << 16));        // dim0[31:16], dim1=1
    g1[3] = (int)((tile_dw & 0xFFFFu) << 16);             // tile_dim0
    g1[4] = 0;                                            // tile_dim1/2 unused
    g1[5] = (int)tensor_dw;                               // dim0_stride (1-D)
    g1[6] = 0;
    g1[7] = 0;
    v4i z4 = (v4i)0;
#if defined(__clang_major__) && __clang_major__ >= 23
    v8i z8 = (v8i)0;
    __builtin_amdgcn_tensor_load_to_lds(g0, g1, z4, z4, z8, 0);
#else
    __builtin_amdgcn_tensor_load_to_lds(g0, g1, z4, z4, 0);
#endif
}

// ---------------------------------------------------------------------------
// Kernel 1: per-atom precompute  za2[n] = {Z[n], Z[n]^|a_exp|}  and zero accum
// ---------------------------------------------------------------------------
__global__ void zbl_init_kernel(const float* __restrict__ Z,
                                const float* __restrict__ a_exp,
                                float2* __restrict__ za2,
                                float* __restrict__ accum, int n) {
    int t = blockIdx.x * blockDim.x + threadIdx.x;
    if (t < n) {
        float ae = fabsf(a_exp[0]);
        float z  = Z[t];
        float za = exp2f(ae * __log2f(z));   // z^ae, z > 0 by construction
        za2[t]   = make_float2(z, za);
        accum[t] = 0.0f;
    }
}

// ---------------------------------------------------------------------------
// Kernel 2: TDM-fed, double-buffered pair kernel.
//   wave 0 issues 4 descriptors for tile k+1, everyone computes tile k,
//   barrier / s_wait_tensorcnt 0 / barrier, swap buffers.
// ---------------------------------------------------------------------------
__global__ void zbl_pairs_kernel(const float* __restrict__ disp,   // [E,3]
                                 const int*   __restrict__ idx_i,  // [E] sorted
                                 const int*   __restrict__ idx_j,  // [E]
                                 const float* __restrict__ bmask,  // [E]
                                 const float2* __restrict__ za2,   // [N] {Z, za}
                                 const float* __restrict__ acoef,  // [1]
                                 const float* __restrict__ pcoef,  // [4]
                                 const float* __restrict__ pexp,   // [4]
                                 float* __restrict__ accum,        // [N]
                                 int E) {
    __shared__ __align__(16) unsigned char ldsbuf[2][BUF_BYTES];

    // ---- per-thread uniform constants: softmax(|c|), |e|, 1/|a| ----
    float a0 = fabsf(pcoef[0]), a1 = fabsf(pcoef[1]);
    float a2 = fabsf(pcoef[2]), a3 = fabsf(pcoef[3]);
    float mc = fmaxf(fmaxf(a0, a1), fmaxf(a2, a3));
    float x0 = __expf(a0 - mc), x1 = __expf(a1 - mc);
    float x2 = __expf(a2 - mc), x3 = __expf(a3 - mc);
    float inv_sum = 1.0f / (x0 + x1 + x2 + x3);
    float c0 = x0 * inv_sum, c1 = x1 * inv_sum, c2 = x2 * inv_sum, c3 = x3 * inv_sum;
    float e0 = fabsf(pexp[0]), e1 = fabsf(pexp[1]);
    float e2 = fabsf(pexp[2]), e3 = fabsf(pexp[3]);
    float inv_a = 1.0f / fabsf(acoef[0]);

    const int tid   = threadIdx.x;
    const int lane  = tid & 31;
    const bool wave0 = (tid < 32);                // uniform per wave
    const int numTiles = (E + TILE - 1) / TILE;
    const unsigned total_disp_dw = (unsigned)E * 3u;

    // prologue: fill buffer 0 with this block's first tile
    int tile = blockIdx.x;
    if (wave0 && tile < numTiles) {
        unsigned base = (unsigned)(size_t)&ldsbuf[0][0];
        unsigned td = (unsigned)tile * (TILE * 3u);
        unsigned te = (unsigned)tile * TILE;
        tdm_load_1d(base,                disp  + td, TILE * 3u, total_disp_dw - td);
        tdm_load_1d(base + DISP_BYTES,   idx_i + te, TILE, (unsigned)E - te);
        tdm_load_1d(base + DISP_BYTES + IDX_BYTES,     idx_j + te, TILE, (unsigned)E - te);
        tdm_load_1d(base + DISP_BYTES + 2 * IDX_BYTES, bmask + te, TILE, (unsigned)E - te);
    }
    if (wave0) __builtin_amdgcn_s_wait_tensorcnt(0);
    __syncthreads();

    int cur = 0;
    for (; tile < numTiles; tile += gridDim.x) {
        // ---- prefetch next tile into the other buffer via TDM ----
        int nxt = tile + gridDim.x;
        if (wave0 && nxt < numTiles) {
            unsigned base = (unsigned)(size_t)&ldsbuf[cur ^ 1][0];
            unsigned td = (unsigned)nxt * (TILE * 3u);
            unsigned te = (unsigned)nxt * TILE;
            tdm_load_1d(base,                disp  + td, TILE * 3u, total_disp_dw - td);
            tdm_load_1d(base + DISP_BYTES,   idx_i + te, TILE, (unsigned)E - te);
            tdm_load_1d(base + DISP_BYTES + IDX_BYTES,     idx_j + te, TILE, (unsigned)E - te);
            tdm_load_1d(base + DISP_BYTES + 2 * IDX_BYTES, bmask + te, TILE, (unsigned)E - te);
        }

        // ---- compute current tile from LDS ----
        const float* dB = (const float*)&ldsbuf[cur][0];
        const int*   iB = (const int*)  &ldsbuf[cur][DISP_BYTES];
        const int*   jB = (const int*)  &ldsbuf[cur][DISP_BYTES + IDX_BYTES];
        const float* mB = (const float*)&ldsbuf[cur][DISP_BYTES + 2 * IDX_BYTES];

        int e = tile * TILE + tid;
        int i = -1;
        float rep = 0.0f;
        if (e < E) {
            float dx = dB[tid * 3 + 0];
            float dy = dB[tid * 3 + 1];
            float dz = dB[tid * 3 + 2];
            i = iB[tid];
            int   j  = jB[tid];
            float bm = mB[tid];
            float2 zi = za2[i];            // L2-resident gathers (800 KB table)
            float2 zj = za2[j];

            float d2   = fmaxf(dx*dx + dy*dy + dz*dz, 1e-20f);
            float rinv = rsqrtf(d2);       // 1/d with d = max(|r|, 1e-10)
            float d    = d2 * rinv;

            float xs   = 5.0f - d;         // switch 1->0 on [4,5]
            float poly = ((6.0f*xs - 15.0f)*xs + 10.0f) * xs * xs * xs;
            float sw   = (d < 4.0f) ? 1.0f : ((d >= 5.0f) ? 0.0f : poly);

            float arg = d * fmaxf(zi.y + zj.y, 1e-10f) * inv_a;
            float t0 = -e0 * arg, t1 = -e1 * arg, t2 = -e2 * arg, t3 = -e3 * arg;
            float m   = fmaxf(fmaxf(t0, t1), fmaxf(t2, t3));   // never added back
            float phi = c0*__expf(t0 - m) + c1*__expf(t1 - m)
                      + c2*__expf(t2 - m) + c3*__expf(t3 - m);
            phi = fmaxf(phi, 1e-30f);
            sw  = fmaxf(sw, 1e-30f);
            rep = 0.5f * zi.x * zj.x * rinv * phi * sw * bm;
        }

        // ---- wave32 segmented inclusive scan keyed on sorted idx_i ----
        #pragma unroll
        for (int off = 1; off < 32; off <<= 1) {
            float r2 = __shfl_up(rep, off, 32);
            int   i2 = __shfl_up(i,   off, 32);
            if (lane >= off && i2 == i) rep += r2;
        }
        int inext = __shfl_down(i, 1, 32);
        if (i >= 0 && (lane == 31 || inext != i))
            atomicAdd(&accum[i], rep);     // ~32x fewer global_atomic_add_f32

        __syncthreads();                               // reads of buf[cur] done
        if (wave0) __builtin_amdgcn_s_wait_tensorcnt(0); // next tile landed
        __syncthreads();                               // buf[cur^1] valid for all
        cur ^= 1;
    }
}

// ---------------------------------------------------------------------------
// Kernel 3: Erep = accum * atom_mask  -> d_out  (shape [N,1,1,1])
// ---------------------------------------------------------------------------
__global__ void zbl_final_kernel(const float* __restrict__ accum,
                                 const float* __restrict__ amask,
                                 float* __restrict__ out, int n) {
    int t = blockIdx.x * blockDim.x + threadIdx.x;
    if (t < n) out[t] = accum[t] * amask[t];
}

// ---------------------------------------------------------------------------
extern "C" void kernel_launch(void* const* d_in, const int* in_sizes, int n_in,
                              void* d_out, int out_size, void* d_ws, size_t ws_size,
                              hipStream_t stream) {
    const float* Z     = (const float*)d_in[0];   // atomic_numbers [N]
    const float* disp  = (const float*)d_in[1];   // displacements [E,3]
    const int*   ii    = (const int*)  d_in[2];   // idx_i [E] (sorted)
    const int*   jj    = (const int*)  d_in[3];   // idx_j [E]
    const float* amask = (const float*)d_in[4];   // atom_mask [N]
    const float* bmask = (const float*)d_in[6];   // batch_mask [E]
    const float* acoef = (const float*)d_in[8];   // a_coefficient [1]
    const float* aexp  = (const float*)d_in[9];   // a_exponent [1]
    const float* pc    = (const float*)d_in[10];  // phi_coefficients [4]
    const float* pe    = (const float*)d_in[11];  // phi_exponents [4]

    const int N = in_sizes[0];
    const int E = in_sizes[2];

    // workspace layout: float2 za2[N] | float accum[N]
    float2* za2   = (float2*)d_ws;
    float*  accum = (float*)((char*)d_ws + (size_t)N * sizeof(float2));

    zbl_init_kernel<<<(N + 255) / 256, 256, 0, stream>>>(Z, aexp, za2, accum, N);

    int numTiles = (E + TILE - 1) / TILE;       // 25000 tiles
    int blocks = numTiles < 2048 ? numTiles : 2048;
    zbl_pairs_kernel<<<blocks, TILE, 0, stream>>>(disp, ii, jj, bmask, za2,
                                                  acoef, pc, pe, accum, E);

    zbl_final_kernel<<<(N + 255) / 256, 256, 0, stream>>>(accum, amask,
                                                          (float*)d_out, N);
}